// Conv_31868657336336
// MI455X (gfx1250) — compile-verified
//
#include <hip/hip_runtime.h>

typedef __attribute__((ext_vector_type(2))) float v2f;
typedef __attribute__((ext_vector_type(8))) float v8f;

#define C_CH 32
#define KVOL 8

// ws layout (32-bit words):
//   [0..7]    counts
//   [8..16]   offsets (padded, row units)
//   [17..24]  cursors
//   [25]      ntiles
//   [32..8223]   Wt: transposed weights, Wt[k][n][c] (8*32*32 floats)
//   [8224..]     order[] (k-sorted input row indices, -1 = padding)
#define WS_COUNTS 0
#define WS_OFFS   8
#define WS_CURS   17
#define WS_NTILES 25
#define WS_WT     32
#define WS_ORDER  (32 + KVOL * C_CH * C_CH)

__global__ void init_kernel(float* out, int out_size, int* ws,
                            const float* __restrict__ W, int order_len) {
    int tid = blockIdx.x * blockDim.x + threadIdx.x;
    if (tid < out_size) out[tid] = 0.0f;
    if (tid < order_len) ws[WS_ORDER + tid] = -1;
    if (tid < KVOL) ws[WS_COUNTS + tid] = 0;
    if (tid < KVOL * C_CH * C_CH) {
        // Wt[k][n][c] = W[k][c][n]  -> B fragments become contiguous float2
        int k = tid >> 10;
        int c = (tid >> 5) & 31;
        int nn = tid & 31;
        float* wt = (float*)(ws + WS_WT);
        wt[(k << 10) + (nn << 5) + c] = W[tid];
    }
}

__global__ void hist_kernel(const int* __restrict__ k_idx, int n, int* ws) {
    int i = blockIdx.x * blockDim.x + threadIdx.x;
    if (i < n) atomicAdd(&ws[WS_COUNTS + k_idx[i]], 1);
}

__global__ void scan_kernel(int* ws) {
    if (threadIdx.x == 0) {
        int off = 0;
        ws[WS_OFFS + 0] = 0;
        for (int k = 0; k < KVOL; ++k) {
            ws[WS_CURS + k] = off;
            int padded = (ws[WS_COUNTS + k] + 15) & ~15;
            off += padded;
            ws[WS_OFFS + k + 1] = off;
        }
        ws[WS_NTILES] = off >> 4;
    }
}

__global__ void scatter_kernel(const int* __restrict__ k_idx, int n, int* ws) {
    int i = blockIdx.x * blockDim.x + threadIdx.x;
    if (i < n) {
        int k = k_idx[i];
        int pos = atomicAdd(&ws[WS_CURS + k], 1);
        ws[WS_ORDER + pos] = i;
    }
}

// One wave = one 16-point tile, all rows sharing one kernel-offset k.
// D = A(16x32) * W[k](32x32) via eight K=4 steps of V_WMMA_F32_16X16X4_F32
// over two N-tiles, then f32 atomic scatter-add into the output voxels.
__global__ void __launch_bounds__(256)
spconv_wmma_kernel(const float* __restrict__ x,
                   const int* __restrict__ out_idx, const int* __restrict__ ws,
                   float* __restrict__ out) {
    const int lane = threadIdx.x & 31;
    const int wave = threadIdx.x >> 5;
    const int tile = blockIdx.x * 8 + wave;
    const int ntiles = ws[WS_NTILES];
    if (tile >= ntiles) return;                 // wave-uniform exit

    const int base = tile * 16;
    // which weight matrix this tile belongs to (bins padded to 16-row multiples)
    int k = 0;
    #pragma unroll
    for (int j = 1; j < KVOL; ++j)
        if (base >= ws[WS_OFFS + j]) k = j;

    const int h = lane >> 4;                    // lane half selects K sub-pair
    const int m = lane & 15;                    // A: row M; B/D: column N
    const int r = ws[WS_ORDER + base + m];      // input row for A loads (-1 = pad)
    const int rc = (r < 0) ? 0 : r;             // clamped: always a valid row
    const float valid = (r < 0) ? 0.0f : 1.0f;  // zero-mask instead of branching
    const float* xrow = x + (long)rc * C_CH;
    const float* Wt   = (const float*)(ws + WS_WT) + (k << 10);
    const float* wrow0 = Wt + (m << 5);         // Wt[k][m][*]      (N-tile 0)
    const float* wrow1 = Wt + ((m + 16) << 5);  // Wt[k][m+16][*]   (N-tile 1)

    v8f c0 = {};  // N = 0..15
    v8f c1 = {};  // N = 16..31

    #pragma unroll
    for (int kk = 0; kk < 8; ++kk) {
        const int K0 = kk * 4 + 2 * h;          // this lane's K pair
        v2f av = *(const v2f*)(xrow + K0);      // unconditional 8B gather
        v2f a;
        a.x = av.x * valid;
        a.y = av.y * valid;
        v2f b0 = *(const v2f*)(wrow0 + K0);     // contiguous in Wt
        v2f b1 = *(const v2f*)(wrow1 + K0);
        c0 = __builtin_amdgcn_wmma_f32_16x16x4_f32(false, a, false, b0, (short)0, c0, false, false);
        c1 = __builtin_amdgcn_wmma_f32_16x16x4_f32(false, a, false, b1, (short)0, c1, false, false);
    }

    // D layout: lane holds (M = j + 8h, N = m) in c0[j] / (N = m+16) in c1[j]
    #pragma unroll
    for (int j = 0; j < 8; ++j) {
        const int M = j + 8 * h;
        const int rr = ws[WS_ORDER + base + M];
        if (rr >= 0) {
            const int o = out_idx[rr];
            atomicAdd(&out[o * C_CH + m],      c0[j]);
            atomicAdd(&out[o * C_CH + 16 + m], c1[j]);
        }
    }
}

extern "C" void kernel_launch(void* const* d_in, const int* in_sizes, int n_in,
                              void* d_out, int out_size, void* d_ws, size_t ws_size,
                              hipStream_t stream) {
    const float* x       = (const float*)d_in[0];
    const float* W       = (const float*)d_in[1];
    const int*   k_idx   = (const int*)d_in[2];
    const int*   out_idx = (const int*)d_in[3];
    float* out = (float*)d_out;
    int*   ws  = (int*)d_ws;

    const int n = in_sizes[0] / C_CH;

    const int order_len = n + KVOL * 16;                 // padded order[] capacity
    int init_n = (out_size > order_len) ? out_size : order_len;
    if (init_n < KVOL * C_CH * C_CH) init_n = KVOL * C_CH * C_CH;
    init_kernel<<<(init_n + 255) / 256, 256, 0, stream>>>(out, out_size, ws, W, order_len);
    hist_kernel<<<(n + 255) / 256, 256, 0, stream>>>(k_idx, n, ws);
    scan_kernel<<<1, 32, 0, stream>>>(ws);
    scatter_kernel<<<(n + 255) / 256, 256, 0, stream>>>(k_idx, n, ws);

    const int tiles_max = (n + KVOL * 15 + 15) / 16;     // upper bound; real count read on-device
    spconv_wmma_kernel<<<(tiles_max + 7) / 8, 256, 0, stream>>>(x, out_idx, ws, out);
}